// ChamferDistanceL2_5248450036647
// MI455X (gfx1250) — compile-verified
//
#include <hip/hip_runtime.h>
#include <hip/hip_bf16.h>

typedef __attribute__((ext_vector_type(2))) float v2f;
typedef __attribute__((ext_vector_type(8))) float v8f;

// One wave owns 32 query columns (two 16-wide B matrices) of cloud Q for one
// batch and streams all of cloud R in 16-row tiles.
// V_WMMA_F32_16X16X4_F32 computes  D[m][n] = ||r_m||^2 - 2 (q_n . r_m)  via
//   A[m] = (rx, ry, rz, ||r||^2)        (rebuilt per R tile, feeds 2 WMMAs)
//   B[n] = (-2qx, -2qy, -2qz, 1)        (loop-invariant, -2 folded here)
// Per-lane D holds 8 rows (the minimized m axis) of one column n, so the min
// over m is 8 in-lane v_min_num ops; lane j / j+16 merge via one xor-shuffle.
__global__ __launch_bounds__(128, 1)
void chamfer_dir_kernel(const float* __restrict__ Q,
                        const float* __restrict__ R,
                        float* __restrict__ out,
                        int nq, int nr,
                        int tilesPerBatch, int totalTiles,
                        float invScale)
{
    const int wave = (int)(threadIdx.x >> 5);
    const int tile = (int)blockIdx.x * (int)(blockDim.x >> 5) + wave;
    if (tile >= totalTiles) return;   // wave-uniform: EXEC stays all-ones

    const int lane = (int)(threadIdx.x & 31);
    const int col  = lane & 15;
    const bool hi  = lane >= 16;      // lanes 16-31 carry the K=2,3 slots

    const int b  = tile / tilesPerBatch;
    const int q0 = (tile - b * tilesPerBatch) << 5;   // 32 query cols per wave

    // Two query points per lane-column: q0+col and q0+col+16.
    const float* qb  = Q + ((size_t)b * (size_t)nq + (size_t)q0) * 3;
    const float* qp0 = qb + (size_t)col * 3;
    const float* qp1 = qb + (size_t)(col + 16) * 3;
    const float q0x = qp0[0], q0y = qp0[1], q0z = qp0[2];
    const float q1x = qp1[0], q1y = qp1[1], q1z = qp1[2];
    const float sqq0 = q0x * q0x + q0y * q0y + q0z * q0z;
    const float sqq1 = q1x * q1x + q1y * q1y + q1z * q1z;

    // B matrices (4x16), -2 folded in, K3 = 1.
    v2f bm0, bm1;
    bm0.x = hi ? (-2.0f * q0z) : (-2.0f * q0x);
    bm0.y = hi ? 1.0f          : (-2.0f * q0y);
    bm1.x = hi ? (-2.0f * q1z) : (-2.0f * q1x);
    bm1.y = hi ? 1.0f          : (-2.0f * q1y);

    const float* rb = R + (size_t)b * (size_t)nr * 3;

    v8f mins0, mins1;
#pragma unroll
    for (int r = 0; r < 8; ++r) { mins0[r] = __builtin_inff(); mins1[r] = __builtin_inff(); }

#pragma unroll 4
    for (int m0 = 0; m0 < nr; m0 += 16) {
        const float* rp = rb + (size_t)(m0 + col) * 3;
        const float rx = rp[0], ry = rp[1], rz = rp[2];
        const float sqr = rx * rx + ry * ry + rz * rz;

        // A matrix (16x4): lanes 0-15 -> (K0,K1)=(x,y); 16-31 -> (K2,K3)=(z,||r||^2)
        v2f am;
        am.x = hi ? rz  : rx;
        am.y = hi ? sqr : ry;

        v8f c = {};  // C = 0 (inline)
        v8f d0 = __builtin_amdgcn_wmma_f32_16x16x4_f32(
            false, am, false, bm0, (short)0, c, false, false);
        v8f d1 = __builtin_amdgcn_wmma_f32_16x16x4_f32(
            false, am, false, bm1, (short)0, c, false, false);

#pragma unroll
        for (int r = 0; r < 8; ++r) {
            mins0[r] = __builtin_fminf(mins0[r], d0[r]);
            mins1[r] = __builtin_fminf(mins1[r], d1[r]);
        }
    }

    // Reduce 8 row-mins per column, then merge lane j (rows 0-7) with j+16 (rows 8-15).
    float c0 = mins0[0], c1 = mins1[0];
#pragma unroll
    for (int r = 1; r < 8; ++r) {
        c0 = __builtin_fminf(c0, mins0[r]);
        c1 = __builtin_fminf(c1, mins1[r]);
    }
    c0 = __builtin_fminf(c0, __shfl_xor(c0, 16, 32));
    c1 = __builtin_fminf(c1, __shfl_xor(c1, 16, 32));

    // dist(q) = ||q||^2 + min_m(||r||^2 - 2 q.r); fold both query columns together.
    float tot = (sqq0 + c0) + (sqq1 + c1);

    // Sum the 16 distinct lane-columns (halves mirrored -> masks 1,2,4,8 suffice).
#pragma unroll
    for (int off = 1; off < 16; off <<= 1)
        tot += __shfl_xor(tot, off, 32);

    if (lane == 0)
        atomicAdd(out + b, tot * invScale);
}

extern "C" void kernel_launch(void* const* d_in, const int* in_sizes, int n_in,
                              void* d_out, int out_size, void* d_ws, size_t ws_size,
                              hipStream_t stream) {
    (void)n_in; (void)d_ws; (void)ws_size;
    const float* xyz1 = (const float*)d_in[0];
    const float* xyz2 = (const float*)d_in[1];
    float* out = (float*)d_out;

    const int B = out_size;                    // [B] output
    const int N = in_sizes[0] / (B * 3);
    const int M = in_sizes[1] / (B * 3);

    // Accumulated via atomics -> must start from zero every call.
    hipMemsetAsync(out, 0, (size_t)out_size * sizeof(float), stream);

    const int wavesPerBlock = 4;               // 128 threads = 4 wave32
    const int threads = wavesPerBlock * 32;

    // Pass 1: queries = xyz1, minimize over xyz2  -> mean(dist1)
    {
        const int tilesPerBatch = N / 32;      // 32 query columns per wave
        const int totalTiles = B * tilesPerBatch;
        const int blocks = (totalTiles + wavesPerBlock - 1) / wavesPerBlock;
        chamfer_dir_kernel<<<blocks, threads, 0, stream>>>(
            xyz1, xyz2, out, N, M, tilesPerBatch, totalTiles, 1.0f / (float)N);
    }
    // Pass 2: queries = xyz2, minimize over xyz1  -> mean(dist2)
    {
        const int tilesPerBatch = M / 32;
        const int totalTiles = B * tilesPerBatch;
        const int blocks = (totalTiles + wavesPerBlock - 1) / wavesPerBlock;
        chamfer_dir_kernel<<<blocks, threads, 0, stream>>>(
            xyz2, xyz1, out, M, N, tilesPerBatch, totalTiles, 1.0f / (float)M);
    }
}